// GCN_seal_1288490189418
// MI455X (gfx1250) — compile-verified
//
#include <hip/hip_runtime.h>

#define H 128
#define PR 144  // padded row stride (float2 units) for swizzled W in LDS: breaks bank conflicts

typedef float v2f __attribute__((ext_vector_type(2)));
typedef float v8f __attribute__((ext_vector_type(8)));

// ---------------- elementwise helpers ----------------

__global__ void k_fill_f32(float* __restrict__ p, float v, int n) {
    int i = blockIdx.x * 256 + threadIdx.x;
    if (i < n) p[i] = v;
}

__global__ void k_count_deg(const int* __restrict__ dst, float* __restrict__ deg, int E) {
    int i = blockIdx.x * 256 + threadIdx.x;
    if (i < E)
        __hip_atomic_fetch_add(&deg[dst[i]], 1.0f, __ATOMIC_RELAXED, __HIP_MEMORY_SCOPE_AGENT);
}

__global__ void k_rsqrt(float* __restrict__ d, int n) {
    int i = blockIdx.x * 256 + threadIdx.x;
    if (i < n) {
        float v = d[i];
        d[i] = (v > 0.0f) ? rsqrtf(v) : 0.0f;
    }
}

__global__ void k_embed(const int* __restrict__ z, const float* __restrict__ tab,
                        float* __restrict__ x, int total) {
    int idx = blockIdx.x * 256 + threadIdx.x;
    if (idx >= total) return;
    int node = idx >> 7;
    int f    = idx & (H - 1);
    x[idx] = tab[(size_t)z[node] * H + f];
}

__global__ void k_relu(float* __restrict__ x, int n) {
    int i = blockIdx.x * 256 + threadIdx.x;
    if (i < n) x[i] = fmaxf(x[i], 0.0f);
}

// ---------------- WMMA f32 GEMM: C[M,128] = A[M,128] @ W[128,128] (+bias)(+relu) ---------
// 256 threads = 8 waves. W is staged into LDS *pre-swizzled* so that a lane's
// B fragment {W[r][n], W[r+1][n]} (r = k + 2*(lane>=16), r even) is one
// contiguous float2 -> single ds_load_b64 per WMMA B operand, no VGPR shuffles.
// Rows padded (PR=144 float2 = +128B) so the two lane-halves hit disjoint
// LDS bank groups. Requires M % 16 == 0 (true here: 100000 and 2000).

__global__ __launch_bounds__(256) void k_gemm_wmma(const float* __restrict__ A,
                                                   const float* __restrict__ W,
                                                   float* __restrict__ C, int M,
                                                   const float* __restrict__ bias,
                                                   int relu) {
    __shared__ v2f Ws2[64 * PR];  // 72 KB of the 320 KB WGP LDS
    {
        int t = threadIdx.x;
#pragma unroll
        for (int j = 0; j < 32; ++j) {          // 64*128 = 8192 pairs / 256 threads
            int idx = t + j * 256;
            int p   = idx >> 7;                 // even-row pair index: rows 2p, 2p+1
            int n   = idx & (H - 1);
            v2f w;
            w.x = W[(2 * p) * H + n];
            w.y = W[(2 * p + 1) * H + n];
            Ws2[p * PR + n] = w;
        }
    }
    __syncthreads();

    int wave    = threadIdx.x >> 5;
    int lane    = threadIdx.x & 31;
    int rowTile = blockIdx.x * 8 + wave;
    if (rowTile * 16 >= M) return;          // whole-wave guard: EXEC stays all-1s for WMMA

    int m0 = rowTile * 16;
    int lh = lane >> 4;                     // 0: K pair {0,1}, 1: K pair {2,3}
    int lm = lane & 15;                     // M (for A) / N (for B,C)

    const float* Arow = A + (size_t)(m0 + lm) * H;

    v8f acc[8] = {};
    for (int k = 0; k < H; k += 4) {
        v2f a = *(const v2f*)(Arow + k + 2 * lh);     // A 16x4 fragment (global_load_b64)
        const v2f* brow = &Ws2[((k >> 1) + lh) * PR + lm];
#pragma unroll
        for (int t = 0; t < 8; ++t) {
            v2f b = brow[t * 16];                     // B 4x16 fragment (ds_load_b64)
            acc[t] = __builtin_amdgcn_wmma_f32_16x16x4_f32(
                false, a, false, b, (short)0, acc[t], false, false);
        }
    }

#pragma unroll
    for (int t = 0; t < 8; ++t) {
#pragma unroll
        for (int v = 0; v < 8; ++v) {
            int row = m0 + v + 8 * lh;              // C/D layout: M = vgpr + 8*(lane>=16)
            int col = t * 16 + lm;
            float val = acc[t][v];
            if (bias) val += bias[col];
            if (relu) val = fmaxf(val, 0.0f);
            C[(size_t)row * H + col] = val;
        }
    }
}

// ---------------- GCN aggregation ----------------

// out[i,:] = convb + h[i,:] * dinv[i]^2   (self-loop term, non-atomic init)
__global__ void k_scatter_init(const float* __restrict__ h, const float* __restrict__ dinv,
                               const float* __restrict__ bias, float* __restrict__ out,
                               int total) {
    int idx = blockIdx.x * 256 + threadIdx.x;
    if (idx >= total) return;
    int node = idx >> 7;
    int f    = idx & (H - 1);
    float di = dinv[node];
    out[idx] = bias[f] + h[idx] * di * di;
}

// one wave per edge, 4 floats per lane: out[dst,:] += h[src,:] * dinv[src]*dinv[dst]
__global__ void k_scatter_edges(const int* __restrict__ src, const int* __restrict__ dst,
                                const float* __restrict__ h, const float* __restrict__ dinv,
                                float* __restrict__ out, int E) {
    int gid  = blockIdx.x * 256 + threadIdx.x;
    int e    = gid >> 5;
    int lane = gid & 31;
    if (e >= E) return;
    int s = src[e];
    int d = dst[e];
    float nrm = dinv[s] * dinv[d];
    float4 hv = *(const float4*)(h + (size_t)s * H + lane * 4);
    float* o  = out + (size_t)d * H + lane * 4;
    __hip_atomic_fetch_add(o + 0, hv.x * nrm, __ATOMIC_RELAXED, __HIP_MEMORY_SCOPE_AGENT);
    __hip_atomic_fetch_add(o + 1, hv.y * nrm, __ATOMIC_RELAXED, __HIP_MEMORY_SCOPE_AGENT);
    __hip_atomic_fetch_add(o + 2, hv.z * nrm, __ATOMIC_RELAXED, __HIP_MEMORY_SCOPE_AGENT);
    __hip_atomic_fetch_add(o + 3, hv.w * nrm, __ATOMIC_RELAXED, __HIP_MEMORY_SCOPE_AGENT);
}

// ---------------- readout ----------------

__global__ void k_center_init(int* __restrict__ c, int G) {
    int i = blockIdx.x * 256 + threadIdx.x;
    if (i < G) c[i] = 0x7fffffff;
}

__global__ void k_center_min(const int* __restrict__ batch, int* __restrict__ c, int N) {
    int i = blockIdx.x * 256 + threadIdx.x;
    if (i < N) atomicMin(&c[batch[i]], i);
}

__global__ void k_pairprod(const int* __restrict__ c, const float* __restrict__ x,
                           float* __restrict__ Hg, int total) {
    int idx = blockIdx.x * 256 + threadIdx.x;
    if (idx >= total) return;
    int g = idx >> 7;
    int f = idx & (H - 1);
    int n = c[g];
    Hg[idx] = x[(size_t)n * H + f] * x[(size_t)(n + 1) * H + f];
}

// one wave per graph: out[g] = dot(T[g,:], W2) + b2
__global__ void k_final(const float* __restrict__ T, const float* __restrict__ W2,
                        const float* __restrict__ b2, float* __restrict__ out, int G) {
    int gid  = blockIdx.x * 256 + threadIdx.x;
    int g    = gid >> 5;
    int lane = gid & 31;
    if (g >= G) return;
    const float* t = T + (size_t)g * H + lane * 4;
    const float* w = W2 + lane * 4;
    float s = t[0] * w[0] + t[1] * w[1] + t[2] * w[2] + t[3] * w[3];
#pragma unroll
    for (int off = 16; off > 0; off >>= 1) s += __shfl_xor(s, off, 32);
    if (lane == 0) out[g] = s + b2[0];
}

// ---------------- launch ----------------

extern "C" void kernel_launch(void* const* d_in, const int* in_sizes, int n_in,
                              void* d_out, int out_size, void* d_ws, size_t ws_size,
                              hipStream_t stream) {
    (void)n_in; (void)ws_size;
    const int*   z     = (const int*)d_in[0];
    const int*   ei    = (const int*)d_in[1];
    const int*   batch = (const int*)d_in[2];
    /* d_in[3] = num_graphs scalar; G known from out_size */
    const float* ztab  = (const float*)d_in[4];
    const float* convW = (const float*)d_in[5];
    const float* convb = (const float*)d_in[6];
    const float* lin1W = (const float*)d_in[7];
    const float* lin1b = (const float*)d_in[8];
    const float* lin2W = (const float*)d_in[9];
    const float* lin2b = (const float*)d_in[10];
    float* out = (float*)d_out;

    int N = in_sizes[0];
    int E = in_sizes[1] / 2;
    int G = out_size;
    const int* esrc = ei;
    const int* edst = ei + E;

    size_t nh   = (size_t)N * H;
    float* ws   = (float*)d_ws;
    float* bufA = ws;                               // x
    float* bufB = ws + nh;                          // h
    float* dinv = ws + 2 * nh;                      // deg -> dinv in place
    float* Hg   = dinv + (((size_t)N + 3) & ~(size_t)3);
    float* T    = Hg + (size_t)G * H;
    int*   cen  = (int*)(T + (size_t)G * H);

    auto blk = [](long n) { return (int)((n + 255) / 256); };

    // degree with self loop, then deg^{-1/2}
    k_fill_f32<<<blk(N), 256, 0, stream>>>(dinv, 1.0f, N);
    k_count_deg<<<blk(E), 256, 0, stream>>>(edst, dinv, E);
    k_rsqrt<<<blk(N), 256, 0, stream>>>(dinv, N);

    // structural embedding gather
    k_embed<<<blk((long)nh), 256, 0, stream>>>(z, ztab, bufA, (int)nh);

    int rowTiles = (N + 15) / 16;
    for (int l = 0; l < 3; ++l) {
        k_gemm_wmma<<<(rowTiles + 7) / 8, 256, 0, stream>>>(
            bufA, convW + (size_t)l * H * H, bufB, N, nullptr, 0);
        k_scatter_init<<<blk((long)nh), 256, 0, stream>>>(bufB, dinv, convb + l * H, bufA, (int)nh);
        k_scatter_edges<<<blk((long)E * 32), 256, 0, stream>>>(esrc, edst, bufB, dinv, bufA, E);
        if (l < 2) k_relu<<<blk((long)nh), 256, 0, stream>>>(bufA, (int)nh);
    }

    // center = first node of each graph; link rep = x[center] * x[center+1]
    k_center_init<<<blk(G), 256, 0, stream>>>(cen, G);
    k_center_min<<<blk(N), 256, 0, stream>>>(batch, cen, N);
    k_pairprod<<<blk((long)G * H), 256, 0, stream>>>(cen, bufA, Hg, G * H);

    // lin1 (bias + relu fused), then lin2 dot
    int rt = (G + 15) / 16;
    k_gemm_wmma<<<(rt + 7) / 8, 256, 0, stream>>>(Hg, lin1W, T, G, lin1b, 1);
    k_final<<<blk((long)G * 32), 256, 0, stream>>>(T, lin2W, lin2b, out, G);
}